// FactorizedReduce_21844203667792
// MI455X (gfx1250) — compile-verified
//
#include <hip/hip_runtime.h>
#include <hip/hip_bf16.h>

typedef __attribute__((ext_vector_type(2))) float v2f;
typedef __attribute__((ext_vector_type(8))) float v8f;

#define C_IN      256
#define C_HALF    128
#define HW_IN     56
#define HW_OUT    28
#define BATCH     32
#define SP_OUT    (HW_OUT * HW_OUT)       // 784
#define N_TOTAL   (BATCH * SP_OUT)        // 25088 per branch
#define NBLK      64                      // N columns per workgroup
#define KCHUNK    16
#define WP        20                      // weight chunk row stride (bank-conflict free)
#define S2        160                     // B pair-row stride; 160 % 64 == 32 -> halves on disjoint banks
#define HWHW      (HW_IN * HW_IN)         // 3136
#define BN_EPS_F  1e-5f

// -------------------------------------------------------------------------
// Kernel 0: zero the stats accumulators (sum[256], sumsq[256]) in workspace.
// -------------------------------------------------------------------------
__global__ void fr_zero_stats(float* __restrict__ stats) {
    int t = threadIdx.x;
    if (t < 512) stats[t] = 0.0f;
}

// -------------------------------------------------------------------------
// Kernel 1: fused ReLU + strided 1x1 conv as WMMA GEMM + stats accumulation.
// grid = (392, 2): x = N-block (64 output positions), y = branch (0: even grid
// with W1, 1: odd grid with W2). 256 threads = 8 wave32.
// Each wave: one 16-row M tile x four 16-col N tiles, K looped in chunks of 16.
// B chunk stored pair-interleaved so each B fragment is one aligned b64 load.
// -------------------------------------------------------------------------
__global__ __launch_bounds__(256)
void fr_gemm_wmma(const float* __restrict__ x,
                  const float* __restrict__ W1,
                  const float* __restrict__ W2,
                  float* __restrict__ out,
                  float* __restrict__ stats) {
    __shared__ float wlds[C_HALF * WP];          // 128 x 16 weight chunk, padded
    __shared__ float blds[(KCHUNK / 2) * S2];    // 8 K-pair rows x 64 n x 2

    const int branch = blockIdx.y;
    const int nblk   = blockIdx.x;
    const int tid    = threadIdx.x;
    const int lane   = tid & 31;
    const int wave   = tid >> 5;           // 0..7
    const int m0     = wave * 16;
    const int hi     = lane >> 4;          // 0 = lanes 0-15, 1 = lanes 16-31
    const int l16    = lane & 15;

    const float* __restrict__ Wg = branch ? W2 : W1;

    // --- per-thread B-load coordinates (fixed n, 4 consecutive k per chunk) ---
    const int n_local = tid & 63;
    const int kq4     = tid >> 6;          // 0..3 -> k = kq4*4 .. kq4*4+3
    const int n_glob  = nblk * NBLK + n_local;
    const int bb      = n_glob / SP_OUT;
    const int sp      = n_glob - bb * SP_OUT;
    const int ho      = sp / HW_OUT;
    const int wo      = sp - ho * HW_OUT;
    const int xbase   = bb * (C_IN * HWHW) + (2 * ho + branch) * HW_IN
                        + (2 * wo + branch);

    // --- weight staging coordinates (2 float4 per thread per chunk) ---
    const int wm = tid >> 1;               // 0..127
    const int wk = (tid & 1) * 8;          // 0 or 8

    v8f acc[4] = {v8f{}, v8f{}, v8f{}, v8f{}};

    for (int kc = 0; kc < C_IN; kc += KCHUNK) {
        __syncthreads();
        // stage weight chunk: Wg[0:128][kc:kc+16] -> wlds (padded rows)
        {
            const float4* src =
                reinterpret_cast<const float4*>(Wg + wm * C_IN + kc + wk);
            float4 w0 = src[0];
            float4 w1 = src[1];
            float4* dst = reinterpret_cast<float4*>(&wlds[wm * WP + wk]);
            dst[0] = w0;
            dst[1] = w1;
        }
        // stage activation chunk with ReLU, fragment-ready pair layout:
        // element (k, n) -> blds[(k>>1)*S2 + 2n + (k&1)]
        {
            const int kb = kq4 * 4;
            float v0 = x[xbase + (kc + kb + 0) * HWHW];
            float v1 = x[xbase + (kc + kb + 1) * HWHW];
            float v2 = x[xbase + (kc + kb + 2) * HWHW];
            float v3 = x[xbase + (kc + kb + 3) * HWHW];
            float2 p0 = make_float2(fmaxf(v0, 0.0f), fmaxf(v1, 0.0f));
            float2 p1 = make_float2(fmaxf(v2, 0.0f), fmaxf(v3, 0.0f));
            *reinterpret_cast<float2*>(&blds[(kq4 * 2 + 0) * S2 + 2 * n_local]) = p0;
            *reinterpret_cast<float2*>(&blds[(kq4 * 2 + 1) * S2 + 2 * n_local]) = p1;
        }
        __syncthreads();

        // compute: 4 K-steps of 4, 4 N tiles -> 16 WMMAs per wave per chunk
        #pragma unroll
        for (int ks = 0; ks < KCHUNK; ks += 4) {
            v2f a;
            {
                const float* ap = &wlds[(m0 + l16) * WP + ks + 2 * hi];
                a.x = ap[0];   // K = ks   (lanes 0-15) / ks+2 (lanes 16-31)
                a.y = ap[1];   // K = ks+1 (lanes 0-15) / ks+3 (lanes 16-31)
            }
            const int prow = (ks >> 1) + hi;   // K-pair row for this half
            #pragma unroll
            for (int t = 0; t < 4; ++t) {
                v2f bf = *reinterpret_cast<const v2f*>(
                    &blds[prow * S2 + 2 * (t * 16 + l16)]);
                acc[t] = __builtin_amdgcn_wmma_f32_16x16x4_f32(
                    false, a, false, bf, (short)0, acc[t], false, false);
            }
        }
    }

    // --- store raw conv output [B, 256, 28, 28] ---
    const int cbase = branch * C_HALF + m0 + 8 * hi;
    #pragma unroll
    for (int t = 0; t < 4; ++t) {
        int ng = nblk * NBLK + t * 16 + l16;
        int ob = ng / SP_OUT;
        int os = ng - ob * SP_OUT;
        float* op = out + ((ob * (2 * C_HALF) + cbase) * SP_OUT + os);
        #pragma unroll
        for (int r = 0; r < 8; ++r) {
            op[r * SP_OUT] = acc[t][r];
        }
    }

    // --- per-channel sum / sumsq: reduce across the 16 lanes of each half ---
    #pragma unroll
    for (int r = 0; r < 8; ++r) {
        float s = acc[0][r] + acc[1][r] + acc[2][r] + acc[3][r];
        float q = acc[0][r] * acc[0][r] + acc[1][r] * acc[1][r]
                + acc[2][r] * acc[2][r] + acc[3][r] * acc[3][r];
        #pragma unroll
        for (int msk = 1; msk < 16; msk <<= 1) {
            s += __shfl_xor(s, msk, 32);
            q += __shfl_xor(q, msk, 32);
        }
        if (l16 == 0) {
            int c = cbase + r;
            atomicAdd(stats + c, s);
            atomicAdd(stats + 256 + c, q);
        }
    }
}

// -------------------------------------------------------------------------
// Kernel 2: turn (sum, sumsq) into per-channel (scale, shift).
// -------------------------------------------------------------------------
__global__ void fr_bn_finalize(const float* __restrict__ gamma,
                               const float* __restrict__ beta,
                               float* __restrict__ stats) {
    int c = threadIdx.x;   // 0..255
    const float inv_n = 1.0f / (float)N_TOTAL;
    float mean = stats[c] * inv_n;
    float var  = stats[256 + c] * inv_n - mean * mean;
    float sc   = gamma[c] * rsqrtf(var + BN_EPS_F);
    stats[512 + c] = sc;
    stats[768 + c] = beta[c] - mean * sc;
}

// -------------------------------------------------------------------------
// Kernel 3: apply BN in place, float4 vectorized (784 % 4 == 0 so each
// float4 lies in a single channel).
// -------------------------------------------------------------------------
__global__ __launch_bounds__(256)
void fr_bn_apply(float* __restrict__ out, const float* __restrict__ stats) {
    int i4 = blockIdx.x * 256 + threadIdx.x;       // float4 index
    int c  = (i4 / (SP_OUT / 4)) & 255;            // (i4/196) % 256
    float sc = stats[512 + c];
    float sh = stats[768 + c];
    float4* p = reinterpret_cast<float4*>(out) + i4;
    float4 v = *p;
    v.x = v.x * sc + sh;
    v.y = v.y * sc + sh;
    v.z = v.z * sc + sh;
    v.w = v.w * sc + sh;
    *p = v;
}

// -------------------------------------------------------------------------
extern "C" void kernel_launch(void* const* d_in, const int* in_sizes, int n_in,
                              void* d_out, int out_size, void* d_ws, size_t ws_size,
                              hipStream_t stream) {
    const float* x     = (const float*)d_in[0];
    const float* W1    = (const float*)d_in[1];
    const float* W2    = (const float*)d_in[2];
    const float* gamma = (const float*)d_in[3];
    const float* beta  = (const float*)d_in[4];
    float* out   = (float*)d_out;
    float* stats = (float*)d_ws;   // [sum:256][sumsq:256][scale:256][shift:256]

    hipLaunchKernelGGL(fr_zero_stats, dim3(1), dim3(512), 0, stream, stats);

    dim3 ggrid(N_TOTAL / NBLK, 2);     // (392, 2)
    hipLaunchKernelGGL(fr_gemm_wmma, ggrid, dim3(256), 0, stream,
                       x, W1, W2, out, stats);

    hipLaunchKernelGGL(fr_bn_finalize, dim3(1), dim3(256), 0, stream,
                       gamma, beta, stats);

    int n4 = (BATCH * 2 * C_HALF * SP_OUT) / 4;    // 1605632
    hipLaunchKernelGGL(fr_bn_apply, dim3(n4 / 256), dim3(256), 0, stream,
                       out, stats);
}